// RowTemporalShiftInterp_7481833030290
// MI455X (gfx1250) — compile-verified
//
#include <hip/hip_runtime.h>
#include <stdint.h>

// Problem constants from the reference: (B,C,T,H,W) = (1,3,3,2160,3840), fp32.
#define W_     3840
#define H_     2160
#define C_     3
#define T_     3
#define BLOCK_ 320                 // 10 wave32s; 3840/320 = 12 cols/thread
#define CPT_   3                   // 16B chunks per thread per row: (3840*4/16)/320
#define COLS_  12                  // output columns per thread

__global__ __launch_bounds__(BLOCK_) void row_temporal_shift_kernel(
    const float* __restrict__ frame,       // (1,3,3,H,W)
    const float* __restrict__ row_shifts,  // (H)
    const float* __restrict__ row_tshifts, // (H/2)
    float* __restrict__ out)               // (1,3,H,W)
{
    const int h   = blockIdx.x;
    const int c   = blockIdx.y;
    const int tid = threadIdx.x;

    __shared__ float lds_cur[W_];
    __shared__ float lds_oth[W_];

    // Row-uniform shift decomposition: floor(col - s) = col + floor(-s).
    const float s = row_shifts[h];
    const int   k = (int)floorf(-s);
    const float w = -s - (float)k;          // bilinear weight, row-constant
    const float w0 = 1.0f - w;

    // Temporal blend: alpha nonzero only on odd rows.
    float a = 0.0f;
    if (h & 1) a = row_tshifts[h >> 1];
    const bool  need_other = (h & 1) != 0;            // block-uniform
    const int   t_other    = (a <= 0.0f) ? 0 : 2;     // prev or next frame
    const float ca         = (a <= 0.0f) ? (1.0f + a) : (1.0f - a);
    const float cb         = (a <= 0.0f) ? (-a)       : a;

    const float* cur_row = frame + (((size_t)c * T_ + 1)       * H_ + h) * (size_t)W_;
    const float* oth_row = frame + (((size_t)c * T_ + t_other) * H_ + h) * (size_t)W_;

    // Low 32 bits of a generic LDS pointer = workgroup-relative LDS byte offset.
    const unsigned lds_cur_base = (unsigned)(uintptr_t)&lds_cur[0];
    const unsigned lds_oth_base = (unsigned)(uintptr_t)&lds_oth[0];

    // Stage the current-frame row into LDS via async 128-bit DMA loads.
    // Rows are single-use -> non-temporal load hint keeps L2 clean.
    #pragma unroll
    for (int i = 0; i < CPT_; ++i) {
        const int chunk = tid + i * BLOCK_;           // 0..959, fully coalesced
        const float* g  = cur_row + chunk * 4;
        const unsigned l = lds_cur_base + (unsigned)(chunk * 16);
        asm volatile("global_load_async_to_lds_b128 %0, %1, off th:TH_LOAD_NT"
                     :: "v"(l), "v"(g) : "memory");
    }
    // Odd rows additionally stage prev or next frame's row.
    if (need_other) {
        #pragma unroll
        for (int i = 0; i < CPT_; ++i) {
            const int chunk = tid + i * BLOCK_;
            const float* g  = oth_row + chunk * 4;
            const unsigned l = lds_oth_base + (unsigned)(chunk * 16);
            asm volatile("global_load_async_to_lds_b128 %0, %1, off th:TH_LOAD_NT"
                         :: "v"(l), "v"(g) : "memory");
        }
    }

    // Drain this wave's async copies, then make them visible workgroup-wide.
    asm volatile("s_wait_asynccnt 0" ::: "memory");
    __syncthreads();

    float* out_row = out + ((size_t)c * H_ + h) * (size_t)W_;

    #pragma unroll
    for (int j = 0; j < COLS_; ++j) {
        const int col = tid + j * BLOCK_;             // coalesced stores
        const int i0  = col + k;
        const int i1  = i0 + 1;
        // Masked taps: OOB contributes 0; LDS loads never fault, so no clamp.
        const float v0 = ((unsigned)i0 < (unsigned)W_) ? lds_cur[i0] : 0.0f;
        const float v1 = ((unsigned)i1 < (unsigned)W_) ? lds_cur[i1] : 0.0f;
        const float cur = __builtin_fmaf(w0, v0, w * v1);

        float val = cur;
        if (need_other) {
            const float u0 = ((unsigned)i0 < (unsigned)W_) ? lds_oth[i0] : 0.0f;
            const float u1 = ((unsigned)i1 < (unsigned)W_) ? lds_oth[i1] : 0.0f;
            const float oth = __builtin_fmaf(w0, u0, w * u1);
            val = __builtin_fmaf(ca, cur, cb * oth);
        }
        // Output is write-once streaming -> non-temporal store.
        __builtin_nontemporal_store(val, out_row + col);
    }
}

extern "C" void kernel_launch(void* const* d_in, const int* in_sizes, int n_in,
                              void* d_out, int out_size, void* d_ws, size_t ws_size,
                              hipStream_t stream) {
    const float* frame       = (const float*)d_in[0];  // (1,3,3,2160,3840) f32
    const float* row_shifts  = (const float*)d_in[1];  // (2160) f32
    const float* row_tshifts = (const float*)d_in[2];  // (1080) f32
    float* out = (float*)d_out;                        // (1,3,2160,3840) f32

    dim3 grid(H_, C_);
    dim3 block(BLOCK_);
    hipLaunchKernelGGL(row_temporal_shift_kernel, grid, block, 0, stream,
                       frame, row_shifts, row_tshifts, out);
}